// SimpleCA_62526133895564
// MI455X (gfx1250) — compile-verified
//
#include <hip/hip_runtime.h>

typedef __attribute__((ext_vector_type(2))) float v2f;
typedef __attribute__((ext_vector_type(8))) float v8f;

// Fixed problem geometry from the reference (B derived at launch).
constexpr int kS   = 512;
constexpr int kSM1 = kS - 1;

// One fused NCA step:
//   perc = depthwise3x3_wrap(x, stencil)          (C=4, per-channel filters)
//   h    = relu(perc @ W1 + b1)                   (4 -> 6)
//   y    = sigmoid(h @ W2)                        (6 -> 4)
//   out  = x*(1-mask) + y*mask
//
// WMMA orientation: pixels ride on the N dimension, tiny weight matrices on M.
//   GEMM1: D1 = A1(W1^T pad 16x4) x B1(perc^T 4x16) + bias  -> lane n owns h[:,n]
//   GEMM2: two K=4 WMMAs implementing K=6 contraction        -> lane n owns y[:,n]
__global__ __launch_bounds__(256) void nca_step_kernel(
    const float* __restrict__ x,     // (B,S,S,4)
    const float* __restrict__ w1,    // (4,6)
    const float* __restrict__ b1,    // (6)
    const float* __restrict__ w2,    // (6,4)
    const float* __restrict__ st,    // (3,3,1,4)
    const float* __restrict__ mask,  // (B,S,S,1)
    float* __restrict__ out)         // (B,S,S,4)
{
    const int tid  = threadIdx.x;
    const int wave = tid >> 5;
    const int lane = tid & 31;
    const int m    = lane & 15;   // M-row index of fragments / pixel-in-tile
    const int hi   = lane >> 4;   // 0 -> K{0,1}, 1 -> K{2,3} half

    const int row = blockIdx.x & kSM1;
    const int b   = blockIdx.x >> 9;   // / kS

    // ---- per-lane constant fragments (loaded once, tiny & cached) ----------
    // A1 = W1^T zero-padded to 16x4: element(m,k) = w1[k*6+m], valid m<6.
    v2f a1;
    a1.x = (m < 6) ? w1[(2 * hi + 0) * 6 + m] : 0.0f;
    a1.y = (m < 6) ? w1[(2 * hi + 1) * 6 + m] : 0.0f;
    // A2 = W2^T zero-padded to 16x8, split into two 16x4 K-slices.
    v2f a2a, a2b;
    a2a.x = (m < 4) ? w2[(2 * hi + 0) * 4 + m] : 0.0f;
    a2a.y = (m < 4) ? w2[(2 * hi + 1) * 4 + m] : 0.0f;
    a2b.x = (m < 4 && hi == 0) ? w2[4 * 4 + m] : 0.0f;   // K=4 row (j=4)
    a2b.y = (m < 4 && hi == 0) ? w2[5 * 4 + m] : 0.0f;   // K=5 row (j=5)
    // C1 = bias broadcast along N for rows M<6, zero elsewhere.
    v8f c1;
#pragma unroll
    for (int r = 0; r < 8; ++r) {
        const int M = r + 8 * hi;
        c1[r] = (M < 6) ? b1[M] : 0.0f;
    }
    // Depthwise stencil weights for this lane's two channels (2*hi, 2*hi+1).
    float wsa[9], wsb[9];
#pragma unroll
    for (int i = 0; i < 9; ++i) {
        wsa[i] = st[i * 4 + 2 * hi + 0];
        wsb[i] = st[i * 4 + 2 * hi + 1];
    }

    const long long imgBase = (long long)b * kS * kS * 4;
    const int rm1 = (row + kS - 1) & kSM1;
    const int rp1 = (row + 1) & kSM1;
    const int rIdx[3] = {rm1, row, rp1};

#pragma unroll
    for (int t = 0; t < 4; ++t) {
        const int col = wave * 64 + t * 16 + m;
        const int cm1 = (col + kS - 1) & kSM1;
        const int cp1 = (col + 1) & kSM1;
        const int cIdx[3] = {cm1, col, cp1};

        // 3x3 wrapped taps; each lane reads a float2 (its two channels).
        float pa = 0.0f, pb = 0.0f;
#pragma unroll
        for (int rr = 0; rr < 3; ++rr) {
#pragma unroll
            for (int cc = 0; cc < 3; ++cc) {
                const float2 v = *(const float2*)(
                    x + imgBase + ((long long)rIdx[rr] * kS + cIdx[cc]) * 4 + 2 * hi);
                const int i = rr * 3 + cc;
                pa = fmaf(wsa[i], v.x, pa);
                pb = fmaf(wsb[i], v.y, pb);
            }
        }
        v2f bperc;
        bperc.x = pa;   // B1 row K = 2*hi
        bperc.y = pb;   // B1 row K = 2*hi + 1

        // GEMM1: H^T = W1^T * perc^T + bias ; then ReLU.
        v8f d1 = __builtin_amdgcn_wmma_f32_16x16x4_f32(
            false, a1, false, bperc, (short)0, c1, false, false);
#pragma unroll
        for (int r = 0; r < 8; ++r) d1[r] = fmaxf(d1[r], 0.0f);

        // Assemble B2 fragments: B2[k][n] = h[k][n].
        // High half-wave needs h rows 2,3 which live in low lanes -> shfl.
        const float h2 = __shfl(d1[2], m, 32);
        const float h3 = __shfl(d1[3], m, 32);
        v2f b2a, b2b;
        b2a.x = hi ? h2 : d1[0];
        b2a.y = hi ? h3 : d1[1];
        b2b.x = hi ? 0.0f : d1[4];
        b2b.y = hi ? 0.0f : d1[5];

        // GEMM2 (K=6 as two chained K=4 WMMAs): Y^T = W2^T * H.
        v8f z = {0.f, 0.f, 0.f, 0.f, 0.f, 0.f, 0.f, 0.f};
        v8f acc = __builtin_amdgcn_wmma_f32_16x16x4_f32(
            false, a2b, false, b2b, (short)0, z, false, false);
        v8f d2 = __builtin_amdgcn_wmma_f32_16x16x4_f32(
            false, a2a, false, b2a, (short)0, acc, false, false);

        // Lane n (low half) now holds y_pre[0..3] of pixel n in d2[0..3].
        if (hi == 0) {
            const long long pix = ((long long)(b * kS + row)) * kS + col;
            const float4 xc = *(const float4*)(x + pix * 4);
            const float  mv = mask[pix];
            // sigmoid via hw exp + fast reciprocal (v_exp_f32 + v_rcp_f32)
            const float y0 = __builtin_amdgcn_rcpf(1.0f + __expf(-d2[0]));
            const float y1 = __builtin_amdgcn_rcpf(1.0f + __expf(-d2[1]));
            const float y2 = __builtin_amdgcn_rcpf(1.0f + __expf(-d2[2]));
            const float y3 = __builtin_amdgcn_rcpf(1.0f + __expf(-d2[3]));
            float4 o;
            o.x = xc.x + mv * (y0 - xc.x);
            o.y = xc.y + mv * (y1 - xc.y);
            o.z = xc.z + mv * (y2 - xc.z);
            o.w = xc.w + mv * (y3 - xc.w);
            *(float4*)(out + pix * 4) = o;
        }
    }
}

extern "C" void kernel_launch(void* const* d_in, const int* in_sizes, int n_in,
                              void* d_out, int out_size, void* d_ws, size_t ws_size,
                              hipStream_t stream) {
    const float* x  = (const float*)d_in[0];
    const float* w1 = (const float*)d_in[1];
    const float* b1 = (const float*)d_in[2];
    const float* w2 = (const float*)d_in[3];
    const float* st = (const float*)d_in[4];
    const float* mk = (const float*)d_in[5];
    float* out = (float*)d_out;

    const int B = in_sizes[0] / (kS * kS * 4);
    dim3 grid(B * kS);   // one block per image row
    dim3 block(256);     // 8 waves x (4 tiles of 16 pixels) = 512-pixel row
    hipLaunchKernelGGL(nca_step_kernel, grid, block, 0, stream,
                       x, w1, b1, w2, st, mk, out);
}